// STDN_5866925326584
// MI455X (gfx1250) — compile-verified
//
#include <hip/hip_runtime.h>

// Bilinear warp (grid-sample) for MI455X / gfx1250.
//
// trace:   (64, 3, 256, 256) fp32   -- gathered (L2-resident, 50 MB total)
// offsets: (64, 2, 256, 256) fp32   -- streamed once (NT loads)
// out:     (64, 3, 256, 256) fp32   -- streamed once (NT stores)
//
// Memory-bound: ~134 MB of compulsory traffic -> ~5.8 us floor at 23.3 TB/s.
// No matmul structure and per-lane data-dependent addresses => WMMA/TDM do not
// apply; the win is weight-reuse across channels + correct temporal hints so
// gather re-reads stay in the 192 MB L2 instead of hitting HBM.

#define PLANE 65536   // 256 * 256
#define WDIM  256

__global__ __launch_bounds__(256) void STDN_warp_bilinear(
    const float* __restrict__ trace,
    const float* __restrict__ offsets,
    float* __restrict__ out,
    int total)            // B * 256 * 256 pixels
{
    int n = blockIdx.x * blockDim.x + threadIdx.x;
    if (n >= total) return;

    int b   = n >> 16;            // batch
    int pix = n & (PLANE - 1);    // i*256 + j within the image
    int i   = pix >> 8;           // H index (first coord, "x" in reference)
    int j   = pix & (WDIM - 1);   // W index (second coord, "y")

    // ---- streaming read of the two offset channels (read-once -> NT) ----
    const float* offb = offsets + (size_t)b * 2 * PLANE + pix;
    float ox = __builtin_nontemporal_load(offb);            // channel 0 -> x
    float oy = __builtin_nontemporal_load(offb + PLANE);    // channel 1 -> y

    // s = clip(grid + off*256, 0, 255)
    float sx = fminf(fmaxf(fmaf(ox, 256.0f, (float)i), 0.0f), 255.0f);
    float sy = fminf(fmaxf(fmaf(oy, 256.0f, (float)j), 0.0f), 255.0f);

    // lu = floor(s), rb = ceil(s)  (ceil, NOT floor+1: integral s degenerates)
    float fx0f = floorf(sx), fy0f = floorf(sy);
    int x0 = (int)fx0f;
    int y0 = (int)fy0f;
    int x1 = (int)ceilf(sx);
    int y1 = (int)ceilf(sy);
    float fx = sx - fx0f;
    float fy = sy - fy0f;

    // flattened gather indices into a 256x256 plane (row = x/H, col = y/W)
    int i00 = (x0 << 8) + y0;   // lu
    int i01 = (x0 << 8) + y1;   // lb
    int i10 = (x1 << 8) + y0;   // ru
    int i11 = (x1 << 8) + y1;   // rb

    const float* tb = trace + (size_t)b * 3 * PLANE;
    float*       ob = out   + (size_t)b * 3 * PLANE + pix;

    // weights computed once, reused for all 3 channels (12 gathers total)
#pragma unroll
    for (int c = 0; c < 3; ++c) {
        const float* tp = tb + c * PLANE;   // RT loads: keep plane hot in L2
        float v00 = tp[i00];
        float v01 = tp[i01];
        float v10 = tp[i10];
        float v11 = tp[i11];

        float xu = fmaf(v10 - v00, fx, v00);   // x_u = lerp(lu, ru, fx)
        float xb = fmaf(v11 - v01, fx, v01);   // x_b = lerp(lb, rb, fx)
        float w  = fmaf(xb - xu, fy, xu);      // lerp(x_u, x_b, fy)

        __builtin_nontemporal_store(w, ob + c * PLANE);  // write-once -> NT
    }
}

extern "C" void kernel_launch(void* const* d_in, const int* in_sizes, int n_in,
                              void* d_out, int out_size, void* d_ws, size_t ws_size,
                              hipStream_t stream) {
    const float* trace   = (const float*)d_in[0];
    const float* offsets = (const float*)d_in[1];
    float*       out     = (float*)d_out;

    // B derived from the offsets tensor: (B, 2, 256, 256)
    int B     = in_sizes[1] / (2 * PLANE);
    int total = B * PLANE;                 // one thread per output pixel

    dim3 block(256);                       // 8 wave32s per block on gfx1250
    dim3 grid((total + block.x - 1) / block.x);
    STDN_warp_bilinear<<<grid, block, 0, stream>>>(trace, offsets, out, total);
}